// ROIRotate_29051158790503
// MI455X (gfx1250) — compile-verified
//
#include <hip/hip_runtime.h>
#include <hip/hip_bf16.h>
#include <cstdint>

// Problem constants (from reference)
#define NB   16
#define CCH  32
#define HF   256
#define WF   256
#define MBOX 1024
#define OH   8
#define OW   128
#define CHSTRIDE (HF * WF)            // 65536 floats between channels (NCHW)
#define OUT_MAIN ((size_t)MBOX * OH * OW * CCH)   // 33,554,432 floats

typedef unsigned int u32;
typedef unsigned long long u64;
typedef __attribute__((ext_vector_type(4))) unsigned int u32x4;
typedef __attribute__((ext_vector_type(8))) int i32x8;
typedef __attribute__((ext_vector_type(4))) int i32x4;

// ---------------------------------------------------------------------------
// Kernel 1: per-box geometry.  params[m*8] = {p1x,p1y,axx,axy,ayx,ayy,(int)width,0}
// ---------------------------------------------------------------------------
__global__ __launch_bounds__(256) void k_setup(const float* __restrict__ boxes,
                                               float* __restrict__ params) {
    int m = blockIdx.x * blockDim.x + threadIdx.x;
    if (m >= MBOX) return;
    const float* b = boxes + m * 8;
    float p1x = b[0] * 0.25f, p1y = b[1] * 0.25f;
    float p2x = b[2] * 0.25f, p2y = b[3] * 0.25f;
    float p4x = b[6] * 0.25f, p4y = b[7] * 0.25f;
    float dwx = p2x - p1x, dwy = p2y - p1y;   // p2 - p1
    float dhx = p4x - p1x, dhy = p4y - p1y;   // p4 - p1
    float box_w = sqrtf(dwx * dwx + dwy * dwy);
    float box_h = sqrtf(dhx * dhx + dhy * dhy);
    bool  swap  = (box_w <= box_h);
    float bw = swap ? box_h : box_w;
    float bh = swap ? box_w : box_h;
    float wreal = ceilf((float)OH * bw / fmaxf(bh, 1e-6f));
    int   width = (int)fminf(fmaxf(wreal, 1.0f), (float)OW);
    float wf = (float)width;
    float* o = params + m * 8;
    o[0] = p1x;  o[1] = p1y;
    o[2] = dwx / wf;      o[3] = dwy / wf;      // ax
    o[4] = dhx * 0.125f;  o[5] = dhy * 0.125f;  // ay = (p4-p1)/HEIGHT
    ((int*)o)[6] = width;
    o[7] = 0.0f;
}

// ---------------------------------------------------------------------------
// Kernel 2: stable argsort by descending width.
// The 1024 widths live strided (one int every 32 B inside params records):
// pull them into LDS with a single TDM strided-tensor DMA (tile 1x1024,
// tensor_dim0_stride = 8 elements, 4 B elements) instead of 1024 scattered
// strided loads, then do the O(M^2) stable rank from LDS.
// ---------------------------------------------------------------------------
__global__ __launch_bounds__(1024) void k_sort(const float* __restrict__ params,
                                               int* __restrict__ order,
                                               float* __restrict__ out_tail) {
    __shared__ int wsh[MBOX];
    const int t = threadIdx.x;

    if (t < 32) {                       // wave 0 issues the DMA
        const u32 lds_addr = (u32)(uintptr_t)wsh;           // LDS offset = addr[31:0]
        const u64 gaddr = (u64)(uintptr_t)params + 24ull;   // &params[0*8+6] (first width)

        // ---- D# group 0 (128b): count=1 | lds_addr | global_addr[56:0] | type=2
        u32x4 g0;
        g0[0] = 1u;                                          // count=1, user mode
        g0[1] = lds_addr;                                    // lds_addr (bytes)
        g0[2] = (u32)(gaddr & 0xffffffffull);                // global_addr[31:0]
        g0[3] = (u32)((gaddr >> 32) & 0x01ffffffull)         // global_addr[56:32]
              | (2u << 30);                                  // type = 2 ("image")

        // ---- D# group 1 (256b)
        i32x8 g1 = {0, 0, 0, 0, 0, 0, 0, 0};
        g1[0] = (int)(2u << 16);        // wg_mask=0 (not in cluster), data_size=2 (4 B)
        g1[1] = (int)(1u    << 16);     // tensor_dim0[15:0]  = 1   (bits 79:48)
        g1[2] = (int)(1024u << 16);     // tensor_dim1[15:0]  = 1024 (bits 111:80)
        g1[3] = (int)(1u    << 16);     // tile_dim0 = 1 (bits 127:112)
        g1[4] = 1024;                   // tile_dim1 = 1024 (bits 143:128)
        g1[5] = 8;                      // tensor_dim0_stride = 8 elements (bits 207:160)

        const i32x4 gz4 = {0, 0, 0, 0};                 // groups 2/3 unused (2-D tensor)
        const i32x8 gz8 = {0, 0, 0, 0, 0, 0, 0, 0};     // extra group (6-arg toolchain form)
        __builtin_amdgcn_tensor_load_to_lds(g0, g1, gz4, gz4, gz8, 0);
        __builtin_amdgcn_s_wait_tensorcnt(0);                // TENSORcnt -> 0
    }
    __syncthreads();

    const int w = wsh[t];
    int rank = 0;
    for (int j = 0; j < MBOX; ++j) {
        const int wj = wsh[j];
        rank += (wj > w) || (wj == w && j < t);  // stable: ties by index
    }
    order[rank] = t;
    out_tail[rank]        = (float)w;   // width[order]
    out_tail[MBOX + rank] = (float)t;   // order
}

// ---------------------------------------------------------------------------
// Kernel 3: main gather + bilinear interp + LDS transpose + coalesced store.
// One block per sorted box. 8 waves (wave32): wave = output row h,
// lane = output column x (32-wide chunks). Inner loop over 32 channels.
// ---------------------------------------------------------------------------
__global__ __launch_bounds__(256) void k_main(const float* __restrict__ fm,
                                              const int* __restrict__ mapping,
                                              const float* __restrict__ params,
                                              const int* __restrict__ order,
                                              float* __restrict__ out) {
    // padded tile: row stride 36 floats (144 B, 16 B aligned) -> float4-safe
    __shared__ float tile[OH][32][36];

    const int mo   = blockIdx.x;
    const int warp = threadIdx.x >> 5;   // 0..7 == output row h
    const int lane = threadIdx.x & 31;

    const int m = order[mo];
    const float* pp = params + m * 8;
    const float p1x = pp[0], p1y = pp[1];
    const float axx = pp[2], axy = pp[3];
    const float ayx = pp[4], ayy = pp[5];
    const int   width = ((const int*)pp)[6];
    const int   n = mapping[m];

    const float* base = fm + (size_t)n * (CCH * CHSTRIDE);
    const float  h = (float)warp;
    const float4 z4 = make_float4(0.f, 0.f, 0.f, 0.f);

    for (int xblk = 0; xblk < 4; ++xblk) {
        const size_t ob = (((size_t)mo * OH + warp) * OW + xblk * 32) * CCH;

        // width is block-uniform: a fully-masked tile needs no gather and no
        // LDS round trip -- store the zeros directly (coalesced b128 rows).
        if (xblk * 32 >= width) {
#pragma unroll
            for (int it = 0; it < 8; ++it)
                ((float4*)(out + ob))[it * 32 + lane] = z4;
            continue;
        }

        const int  x = xblk * 32 + lane;
        const bool colok = (x < width);
        const float fx = (float)x;

        // sample coords
        const float sx = fmaf(axx, fx, fmaf(ayx, h, p1x));
        const float sy = fmaf(axy, fx, fmaf(ayy, h, p1y));
        const float x0f = floorf(sx), y0f = floorf(sy);
        const float tx = sx - x0f, ty = sy - y0f;
        const int x0 = (int)x0f, y0 = (int)y0f;

        // validity folded into weights (reference: v * valid, then col_mask)
        const float mx0 = (x0 >= 0  && x0 < WF)     ? 1.0f : 0.0f;
        const float mx1 = (x0 >= -1 && x0 < WF - 1) ? 1.0f : 0.0f;
        const float my0 = (y0 >= 0  && y0 < HF)     ? 1.0f : 0.0f;
        const float my1 = (y0 >= -1 && y0 < HF - 1) ? 1.0f : 0.0f;
        const float w00 = (1.0f - tx) * (1.0f - ty) * mx0 * my0;
        const float w01 = tx * (1.0f - ty) * mx1 * my0;
        const float w10 = (1.0f - tx) * ty * mx0 * my1;
        const float w11 = tx * ty * mx1 * my1;

        // clamped gather offsets (reused for all 32 channels)
        const int xc0 = min(max(x0, 0), WF - 1);
        const int xc1 = min(max(x0 + 1, 0), WF - 1);
        const int yr0 = min(max(y0, 0), HF - 1) * WF;
        const int yr1 = min(max(y0 + 1, 0), HF - 1) * WF;
        const int off00 = yr0 + xc0, off01 = yr0 + xc1;
        const int off10 = yr1 + xc0, off11 = yr1 + xc1;

        if (colok) {
            // warm L2/L0 for channel 0 rows -> global_prefetch_b8
            __builtin_prefetch(base + off00);
            __builtin_prefetch(base + off10);
            const float* p = base;
#pragma unroll
            for (int c4 = 0; c4 < 8; ++c4) {
                float4 v;
                v.x = w00*p[off00] + w01*p[off01] + w10*p[off10] + w11*p[off11]; p += CHSTRIDE;
                v.y = w00*p[off00] + w01*p[off01] + w10*p[off10] + w11*p[off11]; p += CHSTRIDE;
                v.z = w00*p[off00] + w01*p[off01] + w10*p[off10] + w11*p[off11]; p += CHSTRIDE;
                v.w = w00*p[off00] + w01*p[off01] + w10*p[off10] + w11*p[off11]; p += CHSTRIDE;
                *(float4*)&tile[warp][lane][c4 * 4] = v;   // ds_store_b128
            }
        } else {
#pragma unroll
            for (int c4 = 0; c4 < 8; ++c4)
                *(float4*)&tile[warp][lane][c4 * 4] = z4;
        }

        // intra-wave LDS RAW fence (DS ops are in-order per wave; make the
        // cross-lane visibility explicit and stop compiler reordering)
        asm volatile("s_wait_dscnt 0" ::: "memory");

        // transposed, fully coalesced store: lanes sweep channels.
        // 32x32 floats = 256 float4s, 8 iterations of 32 lanes.
#pragma unroll
        for (int it = 0; it < 8; ++it) {
            const int idx = it * 32 + lane;      // float4 index within tile
            const int r = idx >> 3;              // local x (0..31)
            const int q = idx & 7;               // channel group (0..7)
            float4 v4 = *(const float4*)&tile[warp][r][q * 4];  // ds_load_b128
            ((float4*)(out + ob))[idx] = v4;     // global_store_b128, 128 B/row
        }

        // WAR fence before next xblk reuses the tile
        asm volatile("s_wait_dscnt 0" ::: "memory");
    }
}

// ---------------------------------------------------------------------------
extern "C" void kernel_launch(void* const* d_in, const int* in_sizes, int n_in,
                              void* d_out, int out_size, void* d_ws, size_t ws_size,
                              hipStream_t stream) {
    const float* fm      = (const float*)d_in[0];   // (16,32,256,256) f32
    const float* boxes   = (const float*)d_in[1];   // (1024,8) f32
    const int*   mapping = (const int*)d_in[2];     // (1024,) i32
    float* out = (float*)d_out;

    float* params = (float*)d_ws;                          // 1024*8 floats
    int*   order  = (int*)((char*)d_ws + MBOX * 8 * 4);    // 1024 ints

    k_setup<<<(MBOX + 255) / 256, 256, 0, stream>>>(boxes, params);
    k_sort <<<1, MBOX, 0, stream>>>(params, order, out + OUT_MAIN);
    k_main <<<MBOX, 256, 0, stream>>>(fm, mapping, params, order, out);
}